// GNF_35450660061951
// MI455X (gfx1250) — compile-verified
//
#include <hip/hip_runtime.h>

// GAT normalizing-flow coupling layer for MI455X (gfx1250, wave32).
// Bandwidth-bound: 2 streaming passes over 256MB of int64 edge indices
// (~22us roofline at 23.3 TB/s). Node state (16MB) stays L2-resident;
// edge stream uses non-temporal b128 loads (2 edges/iter) to avoid
// evicting it. Node-level feature transform uses V_WMMA_F32_16X16X4_F32.
//
// Workspace arrays use a padded stride Np = round_up(N, 128) so every
// 16-node WMMA tile stores unconditionally with two b128 stores.

typedef __attribute__((ext_vector_type(2))) float     v2f;
typedef __attribute__((ext_vector_type(8))) float     v8f;
typedef __attribute__((ext_vector_type(2))) long long v2i64;

__device__ __forceinline__ unsigned f2ord(float f) {
  // order-preserving float -> uint mapping (for atomic max)
  unsigned u = __float_as_uint(f);
  return (u & 0x80000000u) ? ~u : (u | 0x80000000u);
}
__device__ __forceinline__ float ord2f(unsigned u) {
  return (u & 0x80000000u) ? __uint_as_float(u & 0x7fffffffu)
                           : __uint_as_float(~u);
}
__device__ __forceinline__ float lrelu(float v) { return v > 0.0f ? v : 0.2f * v; }

// Workspace layout (floats), arrays of stride Np:
//  0: h1_0   1: h1_1   2: as1   3: ad1
//  4: h2_0   5: h2_1   6: as2   7: ad2
//  8: m1 (ordered-uint)   9: m2 (ordered-uint)
// 10: den1  11: num1_0  12: num1_1
// 13: den2  14: num2_0  15: num2_1

// ---------------------------------------------------------------------------
// Kernel 1: fused node transform via WMMA f32 16x16x4.
// Each wave handles 16 nodes: D(16x16) = A(16x4) * B(4x16).
//   A row m = [x0, x1, 0, 0] for node base+m.
//   B col c = coefficients mapping (x0,x1) -> channel c, c in [0,8); cols 8..15 = 0.
// ISA A layout (32-bit, 16x4): lanes 0-15 hold K=0 (vgpr0) / K=1 (vgpr1);
// lanes 16-31 hold K=2/3 (zero here). B symmetric: lanes 0-15 carry K=0/1
// rows for column=lane, lanes 16-31 carry K=2/3 rows (zero).
// D layout: lane c (c<16) holds D[r][c] in vgpr r (r=0..7); lane c+16 holds D[r+8][c].
// Grid covers exactly Np nodes; stores are unconditional into padded arrays.
// ---------------------------------------------------------------------------
__global__ void gat_node_wmma(const float* __restrict__ x,
                              const float* __restrict__ W1, const float* __restrict__ as1,
                              const float* __restrict__ ad1,
                              const float* __restrict__ W2, const float* __restrict__ as2,
                              const float* __restrict__ ad2,
                              float* __restrict__ ws, int N, int Np)
{
  const int lane = threadIdx.x & 31;
  const int wave = threadIdx.x >> 5;
  const int gw   = blockIdx.x * (blockDim.x >> 5) + wave;
  const int base = gw * 16;

  // A matrix fragment
  v2f a; a.x = 0.0f; a.y = 0.0f;
  if (lane < 16) {
    int node = base + lane;
    if (node >= N) node = N - 1;     // clamp: in-bounds read; padded-tail rows
    a.x = x[node * 4 + 0];           // produce unused values only
    a.y = x[node * 4 + 1];
  }

  // B matrix fragment: combined coefficients for 8 output channels.
  //   c0: h1_0  c1: h1_1  c2: as1  c3: ad1  c4: h2_0  c5: h2_1  c6: as2  c7: ad2
  // h[c]  = x0*W[0][c] + x1*W[1][c]          (W row-major 2x2)
  // a_s   = x0*(W00*v0 + W01*v1) + x1*(W10*v0 + W11*v1)
  v2f b; b.x = 0.0f; b.y = 0.0f;
  if (lane < 8) {
    const float* W  = (lane < 4) ? W1  : W2;
    const float* vs = (lane < 4) ? as1 : as2;
    const float* vd = (lane < 4) ? ad1 : ad2;
    const int c = lane & 3;
    if (c == 0)      { b.x = W[0]; b.y = W[2]; }
    else if (c == 1) { b.x = W[1]; b.y = W[3]; }
    else {
      const float* v = (c == 2) ? vs : vd;
      b.x = W[0] * v[0] + W[1] * v[1];
      b.y = W[2] * v[0] + W[3] * v[1];
    }
  }

  v8f acc = {};
  acc = __builtin_amdgcn_wmma_f32_16x16x4_f32(false, a, false, b,
                                              (short)0, acc, false, false);

  // Store: lane (c, half) writes channel c for 8 consecutive nodes.
  // ch*Np is a multiple of 128 floats and nb a multiple of 8 -> 16B aligned.
  const int ch   = lane & 15;
  const int half = lane >> 4;
  if (ch < 8) {
    float* arr = ws + (size_t)ch * Np;
    const int nb = base + half * 8;
    *(float4*)(arr + nb)     = make_float4(acc[0], acc[1], acc[2], acc[3]);
    *(float4*)(arr + nb + 4) = make_float4(acc[4], acc[5], acc[6], acc[7]);
  }
}

// ---------------------------------------------------------------------------
// Kernel 2: per-node init. Seed segment-max with the (always-present)
// self-loop attention value; zero the denominators / numerators.
// ---------------------------------------------------------------------------
__global__ void gat_init(float* __restrict__ ws, int N, int Np)
{
  const int n = blockIdx.x * blockDim.x + threadIdx.x;
  if (n >= N) return;
  const float* as1 = ws + 2 * (size_t)Np;
  const float* ad1 = ws + 3 * (size_t)Np;
  const float* as2 = ws + 6 * (size_t)Np;
  const float* ad2 = ws + 7 * (size_t)Np;
  unsigned* m1 = (unsigned*)(ws + 8 * (size_t)Np);
  unsigned* m2 = (unsigned*)(ws + 9 * (size_t)Np);
  m1[n] = f2ord(lrelu(as1[n] + ad1[n]));
  m2[n] = f2ord(lrelu(as2[n] + ad2[n]));
#pragma unroll
  for (int k = 10; k < 16; ++k) ws[(size_t)k * Np + n] = 0.0f;
}

// ---------------------------------------------------------------------------
// Kernel 3: edge pass 1 — segment max over dst for both convs.
// 2 edges / iteration via 16B non-temporal index loads (global_load_b128).
// ---------------------------------------------------------------------------
__global__ void gat_edge_max(const long long* __restrict__ src,
                             const long long* __restrict__ dst,
                             long long E, float* __restrict__ ws, int Np)
{
  const float* as1 = ws + 2 * (size_t)Np;
  const float* ad1 = ws + 3 * (size_t)Np;
  const float* as2 = ws + 6 * (size_t)Np;
  const float* ad2 = ws + 7 * (size_t)Np;
  unsigned* m1 = (unsigned*)(ws + 8 * (size_t)Np);
  unsigned* m2 = (unsigned*)(ws + 9 * (size_t)Np);

  const long long P   = E >> 1;  // edge pairs
  const long long gid = (long long)blockIdx.x * blockDim.x + threadIdx.x;
  const long long stride = (long long)gridDim.x * blockDim.x;

  for (long long i = gid; i < P; i += stride) {
    const v2i64 sp = __builtin_nontemporal_load((const v2i64*)src + i);
    const v2i64 dp = __builtin_nontemporal_load((const v2i64*)dst + i);
    const int s0 = (int)sp.x, s1 = (int)sp.y;
    const int d0 = (int)dp.x, d1 = (int)dp.y;
    atomicMax(m1 + d0, f2ord(lrelu(as1[s0] + ad1[d0])));
    atomicMax(m2 + d0, f2ord(lrelu(as2[s0] + ad2[d0])));
    atomicMax(m1 + d1, f2ord(lrelu(as1[s1] + ad1[d1])));
    atomicMax(m2 + d1, f2ord(lrelu(as2[s1] + ad2[d1])));
  }
  if ((E & 1) && gid == 0) {               // odd-E tail edge
    const int s = (int)src[E - 1], d = (int)dst[E - 1];
    atomicMax(m1 + d, f2ord(lrelu(as1[s] + ad1[d])));
    atomicMax(m2 + d, f2ord(lrelu(as2[s] + ad2[d])));
  }
}

// ---------------------------------------------------------------------------
// Kernel 4: edge pass 2 — exp(e - m) accumulation of denom and numerators.
// ---------------------------------------------------------------------------
__global__ void gat_edge_acc(const long long* __restrict__ src,
                             const long long* __restrict__ dst,
                             long long E, float* __restrict__ ws, int Np)
{
  const float* h10 = ws + 0 * (size_t)Np;
  const float* h11 = ws + 1 * (size_t)Np;
  const float* as1 = ws + 2 * (size_t)Np;
  const float* ad1 = ws + 3 * (size_t)Np;
  const float* h20 = ws + 4 * (size_t)Np;
  const float* h21 = ws + 5 * (size_t)Np;
  const float* as2 = ws + 6 * (size_t)Np;
  const float* ad2 = ws + 7 * (size_t)Np;
  const unsigned* m1 = (const unsigned*)(ws + 8 * (size_t)Np);
  const unsigned* m2 = (const unsigned*)(ws + 9 * (size_t)Np);
  float* den1 = ws + 10 * (size_t)Np;
  float* n10  = ws + 11 * (size_t)Np;
  float* n11  = ws + 12 * (size_t)Np;
  float* den2 = ws + 13 * (size_t)Np;
  float* n20  = ws + 14 * (size_t)Np;
  float* n21  = ws + 15 * (size_t)Np;

  const long long P   = E >> 1;
  const long long gid = (long long)blockIdx.x * blockDim.x + threadIdx.x;
  const long long stride = (long long)gridDim.x * blockDim.x;

  for (long long i = gid; i < P; i += stride) {
    const v2i64 sp = __builtin_nontemporal_load((const v2i64*)src + i);
    const v2i64 dp = __builtin_nontemporal_load((const v2i64*)dst + i);
#pragma unroll
    for (int k = 0; k < 2; ++k) {
      const int s = (int)(k ? sp.y : sp.x);
      const int d = (int)(k ? dp.y : dp.x);
      const float t1 = __expf(lrelu(as1[s] + ad1[d]) - ord2f(m1[d]));
      atomicAdd(den1 + d, t1);
      atomicAdd(n10 + d, t1 * h10[s]);
      atomicAdd(n11 + d, t1 * h11[s]);
      const float t2 = __expf(lrelu(as2[s] + ad2[d]) - ord2f(m2[d]));
      atomicAdd(den2 + d, t2);
      atomicAdd(n20 + d, t2 * h20[s]);
      atomicAdd(n21 + d, t2 * h21[s]);
    }
  }
  if ((E & 1) && gid == 0) {               // odd-E tail edge
    const int s = (int)src[E - 1], d = (int)dst[E - 1];
    const float t1 = __expf(lrelu(as1[s] + ad1[d]) - ord2f(m1[d]));
    atomicAdd(den1 + d, t1);
    atomicAdd(n10 + d, t1 * h10[s]);
    atomicAdd(n11 + d, t1 * h11[s]);
    const float t2 = __expf(lrelu(as2[s] + ad2[d]) - ord2f(m2[d]));
    atomicAdd(den2 + d, t2);
    atomicAdd(n20 + d, t2 * h20[s]);
    atomicAdd(n21 + d, t2 * h21[s]);
  }
}

// ---------------------------------------------------------------------------
// Kernel 5: finalize. Add self-loop contribution, normalize, bias, then the
// coupling update: x1_new = x2 * exp(s) + t, log_det = s0 + s1.
// d_out = [x1_new (N*2) | x2 (N*2) | log_det (N)]  (float32)
// ---------------------------------------------------------------------------
__global__ void gat_final(const float* __restrict__ x,
                          const float* __restrict__ b1, const float* __restrict__ b2,
                          const float* __restrict__ ws, int N, int Np,
                          float* __restrict__ out)
{
  const int n = blockIdx.x * blockDim.x + threadIdx.x;
  if (n >= N) return;

  const float* h10 = ws + 0 * (size_t)Np;
  const float* h11 = ws + 1 * (size_t)Np;
  const float* as1 = ws + 2 * (size_t)Np;
  const float* ad1 = ws + 3 * (size_t)Np;
  const float* h20 = ws + 4 * (size_t)Np;
  const float* h21 = ws + 5 * (size_t)Np;
  const float* as2 = ws + 6 * (size_t)Np;
  const float* ad2 = ws + 7 * (size_t)Np;
  const unsigned* m1 = (const unsigned*)(ws + 8 * (size_t)Np);
  const unsigned* m2 = (const unsigned*)(ws + 9 * (size_t)Np);
  const float* den1 = ws + 10 * (size_t)Np;
  const float* n10  = ws + 11 * (size_t)Np;
  const float* n11  = ws + 12 * (size_t)Np;
  const float* den2 = ws + 13 * (size_t)Np;
  const float* n20  = ws + 14 * (size_t)Np;
  const float* n21  = ws + 15 * (size_t)Np;

  // F1 -> s (scale)
  const float t1 = __expf(lrelu(as1[n] + ad1[n]) - ord2f(m1[n]));
  const float d1 = den1[n] + t1 + 1e-16f;
  const float s0 = (n10[n] + t1 * h10[n]) / d1 + b1[0];
  const float s1 = (n11[n] + t1 * h11[n]) / d1 + b1[1];

  // F2 -> t (shift)
  const float t2 = __expf(lrelu(as2[n] + ad2[n]) - ord2f(m2[n]));
  const float d2 = den2[n] + t2 + 1e-16f;
  const float u0 = (n20[n] + t2 * h20[n]) / d2 + b2[0];
  const float u1 = (n21[n] + t2 * h21[n]) / d2 + b2[1];

  const float x20 = x[n * 4 + 2];
  const float x21 = x[n * 4 + 3];

  out[2 * (size_t)n + 0] = x20 * __expf(s0) + u0;
  out[2 * (size_t)n + 1] = x21 * __expf(s1) + u1;
  out[2 * (size_t)N + 2 * (size_t)n + 0] = x20;
  out[2 * (size_t)N + 2 * (size_t)n + 1] = x21;
  out[4 * (size_t)N + n] = s0 + s1;
}

extern "C" void kernel_launch(void* const* d_in, const int* in_sizes, int n_in,
                              void* d_out, int out_size, void* d_ws, size_t ws_size,
                              hipStream_t stream)
{
  const float*     x   = (const float*)d_in[0];
  const long long* ei  = (const long long*)d_in[1];
  const float*     W1  = (const float*)d_in[2];
  const float*     as1 = (const float*)d_in[3];
  const float*     ad1 = (const float*)d_in[4];
  const float*     b1  = (const float*)d_in[5];
  const float*     W2  = (const float*)d_in[6];
  const float*     as2 = (const float*)d_in[7];
  const float*     ad2 = (const float*)d_in[8];
  const float*     b2  = (const float*)d_in[9];

  const int       N  = in_sizes[0] / 4;         // 250000
  const int       Np = (N + 127) & ~127;        // padded stride (250112)
  const long long E  = in_sizes[1] / 2;         // 16000000

  float* ws  = (float*)d_ws;                    // needs 16*Np floats ~ 16 MB
  float* out = (float*)d_out;

  const dim3 blk(256);

  // 1) node transform via WMMA (8 waves/block * 16 nodes/wave = 128 nodes/block);
  //    grid covers exactly Np nodes -> unconditional vector stores.
  gat_node_wmma<<<dim3(Np / 128), blk, 0, stream>>>(x, W1, as1, ad1,
                                                    W2, as2, ad2, ws, N, Np);
  // 2) init max with self-loop, zero accumulators
  gat_init<<<dim3((N + 255) / 256), blk, 0, stream>>>(ws, N, Np);
  // 3) edge pass 1: segment max
  gat_edge_max<<<dim3(8192), blk, 0, stream>>>(ei, ei + E, E, ws, Np);
  // 4) edge pass 2: softmax accumulate
  gat_edge_acc<<<dim3(8192), blk, 0, stream>>>(ei, ei + E, E, ws, Np);
  // 5) finalize + coupling update
  gat_final<<<dim3((N + 255) / 256), blk, 0, stream>>>(x, b1, b2, ws, N, Np, out);
}